// FPCELossV2_45251775431143
// MI455X (gfx1250) — compile-verified
//
#include <hip/hip_runtime.h>
#include <hip/hip_bf16.h>
#include <stdint.h>

#define CCLS 16
#define TP   256                  // positions per tile
#define TILE_FLOATS (CCLS * TP)   // 4096 floats = 16 KB

// ---------- fast math helpers (hardware exp2/log2) ----------
__device__ __forceinline__ float fast_exp2(float x) {
#if __has_builtin(__builtin_amdgcn_exp2f)
  return __builtin_amdgcn_exp2f(x);
#else
  return exp2f(x);
#endif
}
__device__ __forceinline__ float fast_log2(float x) {
#if __has_builtin(__builtin_amdgcn_logf)
  return __builtin_amdgcn_logf(x);
#else
  return __log2f(x);
#endif
}

// ---------- kernel 1: zero scratch accumulators ----------
__global__ void fpce_init(unsigned* __restrict__ freq, float* __restrict__ acc) {
  const int t = threadIdx.x;
  if (t < CCLS) freq[t] = 0u;
  if (t < 2)    acc[t]  = 0.0f;
}

// ---------- kernel 2: label histogram via wave32 ballot ----------
__global__ __launch_bounds__(256) void fpce_hist(const long long* __restrict__ lab,
                                                 unsigned* __restrict__ freq, int n) {
  unsigned cnt = 0;
  const int lane   = threadIdx.x & 31;
  const int stride = gridDim.x * blockDim.x;
  for (int i = blockIdx.x * blockDim.x + threadIdx.x; i < n; i += stride) {
    const int v = (int)lab[i];
#pragma unroll
    for (int c = 0; c < CCLS; ++c) {
      unsigned long long m = __ballot(v == c);
      if (lane == c) cnt += (unsigned)__popcll(m);
    }
  }
  if (lane < CCLS) atomicAdd(&freq[lane], cnt);
}

// ---------- kernel 3: class weights ----------
__global__ void fpce_weight(const unsigned* __restrict__ freq,
                            float* __restrict__ w, int n) {
  const int c = threadIdx.x;
  if (c < CCLS) {
    const float ratio = (float)freq[c] / (float)n;
    w[c] = 0.3f + 0.3f / (ratio + 0.02f) + 0.4f / ratio;
  }
}

// =====================================================================
// Tile staging: Tensor Data Mover (preferred) or async-to-LDS fallback
// =====================================================================
#if __has_builtin(__builtin_amdgcn_tensor_load_to_lds)
#define HAVE_TDM 1
typedef unsigned int v4u __attribute__((ext_vector_type(4)));
typedef int          v8i __attribute__((ext_vector_type(8)));
typedef int          v4i __attribute__((ext_vector_type(4)));

// One TDM descriptor moves a 16x256 f32 tile (rows strided by N floats)
// from global into LDS.  Issued once per wave; tracked with TENSORcnt.
__device__ __forceinline__ void tdm_tile_load(uint64_t gaddr, uint32_t lds_addr, int N) {
  v4u g0;
  g0[0] = 1u;                                           // count=1 valid, no gather
  g0[1] = lds_addr;                                     // LDS byte address
  g0[2] = (unsigned)(gaddr & 0xFFFFFFFFull);            // global_addr[31:0]
  g0[3] = (unsigned)((gaddr >> 32) & 0x01FFFFFFull)     // global_addr[56:32]
        | (2u << 30);                                   // type = 2 ("image")
  v8i g1;
  g1[0] = 0x00020000;                                   // data_size = 4 bytes
  g1[1] = (int)(((unsigned)N & 0xFFFFu) << 16);         // tensor_dim0[15:0]
  g1[2] = (int)((((unsigned)N) >> 16) | (16u << 16));   // tensor_dim0[31:16] | tensor_dim1=16
  g1[3] = (int)(256u << 16);                            // tensor_dim1 hi=0 | tile_dim0=256
  g1[4] = 16;                                           // tile_dim1=16, tile_dim2=0
  g1[5] = N;                                            // tensor_dim0_stride[31:0]
  g1[6] = 0;                                            // stride hi / dim1_stride lo
  g1[7] = 0;
  v4i gz = {0, 0, 0, 0};                                // 2D: groups 2/3 unused
#if defined(__clang_major__) && (__clang_major__ >= 23)
  v8i gz8 = {0, 0, 0, 0, 0, 0, 0, 0};
  __builtin_amdgcn_tensor_load_to_lds(g0, g1, gz, gz, gz8, 0);
#else
  __builtin_amdgcn_tensor_load_to_lds(g0, g1, gz, gz, 0);
#endif
}
#else
#define HAVE_TDM 0
// Fallback: per-lane async copies (ASYNCcnt-tracked), 4 x B128 per thread.
__device__ __forceinline__ void tile_prefetch_async(uint64_t pred_base, int N, int tile,
                                                    uint32_t lds_base, int tid) {
  const uint32_t p0 = (uint32_t)tile * TP;
#pragma unroll
  for (int j = 0; j < 4; ++j) {
    const uint32_t k    = (uint32_t)tid + 256u * (uint32_t)j;
    const uint32_t c    = k >> 6;
    const uint32_t i4   = (k & 63u) << 2;
    const uint32_t goff = (c * (uint32_t)N + p0 + i4) * 4u;
    const uint32_t doff = lds_base + (c * TP + i4) * 4u;
    asm volatile("global_load_async_to_lds_b128 %0, %1, %2"
                 :
                 : "v"(doff), "v"(goff), "s"(pred_base)
                 : "memory");
  }
}
#endif

// ---------- kernel 4: main streaming loss pass ----------
__global__ __launch_bounds__(256) void fpce_main(const float* __restrict__ pred,
                                                 const long long* __restrict__ lab,
                                                 const float* __restrict__ w,
                                                 float* __restrict__ acc, int N) {
  __shared__ __align__(16) float buf[2][TILE_FLOATS];   // 32 KB double buffer
  __shared__ float wsh[CCLS];

  const int tid = threadIdx.x;
  if (tid < CCLS) wsh[tid] = w[tid];

  const uint64_t pbase = (uint64_t)(uintptr_t)pred;
  // ISA: flat LDS addresses keep the group-relative offset in addr[31:0]
  const uint32_t lds0 = (uint32_t)(uintptr_t)(&buf[0][0]);
  const uint32_t lds1 = (uint32_t)(uintptr_t)(&buf[1][0]);

  const int ntiles = N / TP;
  const int stride = gridDim.x;

  float lacc = 0.0f, wacc = 0.0f;
  int parity = 0;

#if HAVE_TDM
  const bool issuer = (tid < 32);                       // wave 0 drives the DMA
  if (issuer && (int)blockIdx.x < ntiles)
    tdm_tile_load(pbase + (uint64_t)blockIdx.x * (TP * 4u), lds0, N);
#else
  if ((int)blockIdx.x < ntiles)
    tile_prefetch_async(pbase, N, blockIdx.x, lds0, tid);
#endif

  for (int t = blockIdx.x; t < ntiles; t += stride) {
    const int nt = t + stride;
#if HAVE_TDM
    if (issuer) {
      if (nt < ntiles) {
        tdm_tile_load(pbase + (uint64_t)nt * (TP * 4u), parity ? lds0 : lds1, N);
        __builtin_amdgcn_s_wait_tensorcnt(1);           // retire tile t, keep t+stride in flight
      } else {
        __builtin_amdgcn_s_wait_tensorcnt(0);
      }
    }
#else
    if (nt < ntiles) {
      tile_prefetch_async(pbase, N, nt, parity ? lds0 : lds1, tid);
      asm volatile("s_wait_asynccnt 4" ::: "memory");
    } else {
      asm volatile("s_wait_asynccnt 0" ::: "memory");
    }
#endif
    __syncthreads();                                    // data visible to all 8 waves

    const float* tb = parity ? &buf[1][0] : &buf[0][0];
    const int p = t * TP + tid;

    float xs[CCLS];
#pragma unroll
    for (int c = 0; c < CCLS; ++c) xs[c] = tb[c * TP + tid];  // stride-256: conflict-free

    float m = xs[0];
    int am = 0;
#pragma unroll
    for (int c = 1; c < CCLS; ++c) {
      if (xs[c] > m) { m = xs[c]; am = c; }             // first-max ties, matches argmax
    }
    float s = 0.0f;
#pragma unroll
    for (int c = 0; c < CCLS; ++c)
      s += fast_exp2((xs[c] - m) * 1.44269504088896340736f);
    const float lse = m + 0.69314718055994530942f * fast_log2(s);

    const int ti = (int)lab[p];
    const float wv = (ti == 0) ? wsh[am] : wsh[ti];
    lacc += wv * (lse - xs[ti]);                        // == -wv * logp_true
    wacc += wv;

    __syncthreads();                                    // buffer safe to overwrite
    parity ^= 1;
  }

  // wave32 reduction, then 2 global atomics per wave
#pragma unroll
  for (int off = 16; off > 0; off >>= 1) {
    lacc += __shfl_down(lacc, off);
    wacc += __shfl_down(wacc, off);
  }
  if ((tid & 31) == 0) {
    atomicAdd(&acc[0], lacc);
    atomicAdd(&acc[1], wacc);
  }
}

// ---------- kernel 5: finalize ----------
__global__ void fpce_fin(const float* __restrict__ acc, float* __restrict__ out) {
  out[0] = acc[0] / acc[1];
}

extern "C" void kernel_launch(void* const* d_in, const int* in_sizes, int n_in,
                              void* d_out, int out_size, void* d_ws, size_t ws_size,
                              hipStream_t stream) {
  const float*     pred = (const float*)d_in[0];
  const long long* lab  = (const long long*)d_in[1];   // int64 labels
  float* out = (float*)d_out;

  const int N = in_sizes[1];       // 4,194,304 positions

  // d_ws layout: [0..15] unsigned freq | [16..31] float w | [32..33] float acc
  unsigned* freq = (unsigned*)d_ws;
  float*    w    = (float*)d_ws + 16;
  float*    acc  = (float*)d_ws + 32;

  fpce_init<<<1, 32, 0, stream>>>(freq, acc);

  fpce_hist<<<512, 256, 0, stream>>>(lab, freq, N);

  fpce_weight<<<1, CCLS, 0, stream>>>(freq, w, N);

  const int ntiles = N / TP;
  int grid = ntiles < 4096 ? ntiles : 4096;
  if (grid < 1) grid = 1;
  fpce_main<<<grid, 256, 0, stream>>>(pred, lab, w, acc, N);

  fpce_fin<<<1, 1, 0, stream>>>(acc, out);
}